// S6_55207509622813
// MI455X (gfx1250) — compile-verified
//
#include <hip/hip_runtime.h>

typedef __attribute__((ext_vector_type(2))) float v2f;
typedef __attribute__((ext_vector_type(8))) float v8f;

#define BATCH 2
#define SEQ   2048
#define DIM   1024
#define HID   16
#define MROWS (BATCH * SEQ)            // 4096 flattened (b,l) rows
#define MTILES (MROWS / 16)            // 256 WMMA row-tiles
#define WAVES_A 4                      // waves per block (split-K) in proj kernel
#define KSLICE (DIM / WAVES_A)         // 256 K per wave

#define TCHUNK 64                      // scan timesteps staged per LDS buffer
#define CHUNKS (SEQ / TCHUNK)          // 32

// ---- CDNA5 async global->LDS copy helpers (ASYNCcnt-tracked) ---------------
__device__ __forceinline__ void async_ld_b128(void* lds, const void* gsrc) {
    uint32_t l = (uint32_t)(uintptr_t)lds;      // low 32 bits = LDS byte addr
    asm volatile("global_load_async_to_lds_b128 %0, %1, off"
                 :: "v"(l), "v"(gsrc) : "memory");
}
__device__ __forceinline__ void async_ld_b64(void* lds, const void* gsrc) {
    uint32_t l = (uint32_t)(uintptr_t)lds;
    asm volatile("global_load_async_to_lds_b64 %0, %1, off"
                 :: "v"(l), "v"(gsrc) : "memory");
}
#define ASYNC_WAIT(n) asm volatile("s_wait_asynccnt " #n ::: "memory")

// ---------------------------------------------------------------------------
// Kernel 1: fused projections via V_WMMA_F32_16X16X4_F32, split-K x4.
//   Bm = x @ W_B + bB ; Cm = x @ W_C + bC ; Sd = x @ W_delta + bD
// One block owns a 16-row tile; wave w covers K in [w*256,(w+1)*256).
// A-frag (16x4 f32): lane(0-15)=rows K{0,1}; lane(16-31)=rows K{2,3}.
// B-frag (4x16 f32): lane&15 = N, (lane>>4)*2 = K base.
// ---------------------------------------------------------------------------
__global__ __launch_bounds__(WAVES_A * 32)
void s6_proj_wmma(const float* __restrict__ x,    // [4096,1024]
                  const float* __restrict__ Wb,   // [1024,16]
                  const float* __restrict__ bB,   // [16]
                  const float* __restrict__ Wc,   // [1024,16]
                  const float* __restrict__ bC,   // [16]
                  const float* __restrict__ Wd,   // [1024]
                  const float* __restrict__ bD,   // [1]
                  float* __restrict__ Bm,         // [4096,16]
                  float* __restrict__ Cm,         // [4096,16]
                  float* __restrict__ Sd)         // [4096]
{
    __shared__ float redB[WAVES_A][16][16];
    __shared__ float redC[WAVES_A][16][16];
    __shared__ float redS[WAVES_A][16];

    const int lane = threadIdx.x & 31;
    const int wave = threadIdx.x >> 5;
    const int m0   = blockIdx.x * 16;
    const int row  = m0 + (lane & 15);
    const int kb   = (lane >> 4) * 2;               // 0 or 2
    const int n    = lane & 15;
    const int kbase = wave * KSLICE;

    v8f accB = {};
    v8f accC = {};
    float ds = 0.0f;

    const float* xrow = x + (size_t)row * DIM + kbase;

    #pragma unroll 4
    for (int k0 = 0; k0 < KSLICE; k0 += 4) {
        const int kk = kbase + k0 + kb;
        v2f a = *(const v2f*)(xrow + k0 + kb);          // 2 consecutive K of x
        v2f bb, bc;
        bb.x = Wb[kk * HID + n];
        bb.y = Wb[(kk + 1) * HID + n];
        bc.x = Wc[kk * HID + n];
        bc.y = Wc[(kk + 1) * HID + n];
        v2f wd = *(const v2f*)(Wd + kk);

        accB = __builtin_amdgcn_wmma_f32_16x16x4_f32(false, a, false, bb,
                                                     (short)0, accB, false, false);
        accC = __builtin_amdgcn_wmma_f32_16x16x4_f32(false, a, false, bc,
                                                     (short)0, accC, false, false);
        ds += a.x * wd.x + a.y * wd.y;                  // N=1 delta projection
    }

    // Dump per-wave partial tiles. D layout: VGPR v -> row v + 8*(lane>>4).
    const int rrow = (lane >> 4) << 3;
    #pragma unroll
    for (int v = 0; v < 8; ++v) {
        redB[wave][rrow + v][n] = accB[v];
        redC[wave][rrow + v][n] = accC[v];
    }
    ds += __shfl_xor(ds, 16, 32);       // pair lane L with L^16: full K-slice
    if (lane < 16) redS[wave][lane] = ds;
    __syncthreads();

    // Cross-wave reduction + bias + store (128 threads, 256 elems/matrix).
    for (int i = threadIdx.x; i < 256; i += WAVES_A * 32) {
        const int m = i >> 4, c = i & 15;
        float sB = redB[0][m][c] + redB[1][m][c] + redB[2][m][c] + redB[3][m][c];
        float sC = redC[0][m][c] + redC[1][m][c] + redC[2][m][c] + redC[3][m][c];
        Bm[(size_t)(m0 + m) * HID + c] = sB + bB[c];
        Cm[(size_t)(m0 + m) * HID + c] = sC + bC[c];
    }
    if (threadIdx.x < 16) {
        Sd[m0 + threadIdx.x] = redS[0][threadIdx.x] + redS[1][threadIdx.x] +
                               redS[2][threadIdx.x] + redS[3][threadIdx.x] + bD[0];
    }
}

// ---------------------------------------------------------------------------
// Kernel 2: selective scan, 1 lane per (d,n) chain. ALL per-step operands
// (x-slice, Bm, Cm, Sd) staged in LDS via double-buffered
// GLOBAL_LOAD_ASYNC_TO_LDS (ASYNCcnt-fenced) -> hot loop is 100% LDS-fed.
//   lane = d_sub*16 + n ; wave covers 2 d x 16 n ; block covers 16 d.
//   grid = BATCH * (DIM/16) = 128 blocks x 8 waves = 1024 waves.
// Per chunk each wave issues exactly 3 b128 asyncs (x,Bm,Cm); wave 0 adds one
// b64 for Sd -> per-wave in-flight counts are wave-constant (waits 4/3/0).
// ---------------------------------------------------------------------------
__global__ __launch_bounds__(256)
void s6_scan(const float* __restrict__ x,     // [2,2048,1024]
             const float* __restrict__ A,     // [1024,16]
             const float* __restrict__ dparm, // [1024]
             const float* __restrict__ Bm,    // [4096,16]
             const float* __restrict__ Cm,    // [4096,16]
             const float* __restrict__ Sd,    // [4096]
             float* __restrict__ y)           // [2,2048,1024]
{
    __shared__ __align__(16) float shX[2][TCHUNK * 16];   // 2 x 4KB (16 d's)
    __shared__ __align__(16) float shB[2][TCHUNK * HID];  // 2 x 4KB
    __shared__ __align__(16) float shC[2][TCHUNK * HID];  // 2 x 4KB
    __shared__ __align__(16) float shS[2][TCHUNK];        // 2 x 256B

    const int lane = threadIdx.x & 31;
    const int wave = threadIdx.x >> 5;
    const int b    = blockIdx.x >> 6;           // 64 blocks per batch
    const int d0   = (blockIdx.x & 63) * 16;
    const int dd   = wave * 2 + (lane >> 4);    // 0..15 within block's d-slice
    const int d    = d0 + dd;
    const int n    = lane & 15;

    const float Adn = A[d * HID + n];
    const float dpv = dparm[d];
    float h = 0.0f;

    const char* Xg = (const char*)(x + (size_t)b * SEQ * DIM + d0);
    const char* Bg = (const char*)(Bm + (size_t)b * SEQ * HID);
    const char* Cg = (const char*)(Cm + (size_t)b * SEQ * HID);
    const char* Sg = (const char*)(Sd + (size_t)b * SEQ);
    float*      yb = y + (size_t)b * SEQ * DIM + d;

    // Bm/Cm chunk = contiguous 4KB: wave w copies 512B, lane l 16B of it.
    const uint32_t woff = (uint32_t)(wave * 512 + lane * 16);
    // x chunk = 64 strided rows of 64B: wave w copies rows w*8..w*8+7,
    // 4 lanes x 16B per row.
    const int xrow  = wave * 8 + (lane >> 2);          // row within chunk
    const int xboff = (lane & 3) * 16;                 // byte offset in row

    // -- prefetch issue: MUST be executed uniformly by every wave ----------
    auto prefetch = [&](int buf, int c) {
        const size_t cb = (size_t)c * TCHUNK * HID * sizeof(float); // 4KB
        async_ld_b128((char*)&shX[buf][0] + xrow * 64 + xboff,
                      Xg + (size_t)(c * TCHUNK + xrow) * (DIM * sizeof(float)) + xboff);
        async_ld_b128((char*)&shB[buf][0] + woff, Bg + cb + woff);
        async_ld_b128((char*)&shC[buf][0] + woff, Cg + cb + woff);
        if (wave == 0)
            async_ld_b64((char*)&shS[buf][0] + lane * 8,
                         Sg + (size_t)c * TCHUNK * sizeof(float) + lane * 8);
    };

    prefetch(0, 0);
    prefetch(1, 1);

    for (int c = 0; c < CHUNKS; ++c) {
        const int buf = c & 1;
        // Wait for chunk c's own asyncs (in-order completion): leave only the
        // next chunk's (3 or 4) outstanding.
        if (c + 1 < CHUNKS) {
            if (wave == 0) { ASYNC_WAIT(4); } else { ASYNC_WAIT(3); }
        } else {
            ASYNC_WAIT(0);
        }
        __syncthreads();                     // all waves' copies visible

        const int t0 = c * TCHUNK;
        #pragma unroll 4
        for (int tt = 0; tt < TCHUNK; ++tt) {
            // delta = softplus(delta_param[d] + s_delta[b,t])  (stable form)
            float z     = dpv + shS[buf][tt];
            float delta = fmaxf(z, 0.0f) + __logf(1.0f + __expf(-fabsf(z)));
            float a     = __expf(delta * Adn);          // A_disc (ZOH)
            float xv    = shX[buf][tt * 16 + dd];
            float Bv    = shB[buf][tt * HID + n];
            float Cv    = shC[buf][tt * HID + n];

            h = a * h + delta * Bv * xv;                // h_t = A*h + (dB)*x
            float p = h * Cv;                           // partial <h, C>
            p += __shfl_xor(p, 8, 32);                  // reduce over n
            p += __shfl_xor(p, 4, 32);
            p += __shfl_xor(p, 2, 32);
            p += __shfl_xor(p, 1, 32);
            if (n == 0) yb[(size_t)(t0 + tt) * DIM] = p;
        }
        __syncthreads();                     // done reading buf before reuse
        if (c + 2 < CHUNKS) prefetch(buf, c + 2);
    }
}

// ---------------------------------------------------------------------------
extern "C" void kernel_launch(void* const* d_in, const int* in_sizes, int n_in,
                              void* d_out, int out_size, void* d_ws, size_t ws_size,
                              hipStream_t stream) {
    const float* x   = (const float*)d_in[0];
    const float* A   = (const float*)d_in[1];
    const float* Wb  = (const float*)d_in[2];
    const float* bB  = (const float*)d_in[3];
    const float* Wc  = (const float*)d_in[4];
    const float* bC  = (const float*)d_in[5];
    const float* dpm = (const float*)d_in[6];
    const float* Wd  = (const float*)d_in[7];
    const float* bD  = (const float*)d_in[8];
    float* y = (float*)d_out;

    float* Bm = (float*)d_ws;                 // [4096,16]
    float* Cm = Bm + (size_t)MROWS * HID;     // [4096,16]
    float* Sd = Cm + (size_t)MROWS * HID;     // [4096]

    s6_proj_wmma<<<MTILES, WAVES_A * 32, 0, stream>>>(
        x, Wb, bB, Wc, bC, Wd, bD, Bm, Cm, Sd);

    s6_scan<<<BATCH * (DIM / 16), 256, 0, stream>>>(
        x, A, dpm, Bm, Cm, Sd, y);
}